// HardPhongShader_23295902613849
// MI455X (gfx1250) — compile-verified
//
#include <hip/hip_runtime.h>
#include <stdint.h>

// ---------------------------------------------------------------------------
// HardPhongShader for MI455X (gfx1250)
//
// Memory-bound gather/stream kernel. CDNA5-specific data movement:
//   * TENSOR_LOAD_TO_LDS (TDM) double-buffered staging of the streaming
//     inputs (pix_to_face + bary), one private 32-pixel slice per wave ->
//     no workgroup barriers, sync purely via s_wait_tensorcnt.
//   * Non-temporal b128 RGBA stores (keep 192MB L2 for the gather tables).
// Gather tables (verts/normals/colors 3.6MB, faces 2.4MB) are L2-resident.
// ---------------------------------------------------------------------------

typedef __attribute__((ext_vector_type(4))) float        v4f;
typedef __attribute__((ext_vector_type(4))) int          v4i;
typedef __attribute__((ext_vector_type(4))) unsigned int v4u;
typedef __attribute__((ext_vector_type(8))) int          v8i;

#define TILE            256   // pixels per tile (== blockDim.x)
#define TILES_PER_BLOCK 4     // tiles per block (double-buffered)
#define LANES           32    // wave32

// Low 32 bits of a generic LDS pointer are the workgroup-relative byte
// offset (aperture bits live in the high dword per the CDNA5 aperture check).
__device__ __forceinline__ unsigned lds_byte_off(const void* p) {
  return (unsigned)(uintptr_t)p;
}

// Issue one TDM 1-D copy: elems_f32 DWORDs from global -> LDS.
// D# group layout per CDNA5 ISA ch. 7/8 (CDNA5 ISA).
__device__ __forceinline__ void tdm_load_1d(unsigned lds_addr,
                                            const void* gptr,
                                            unsigned elems_f32) {
  unsigned long long ga = (unsigned long long)(uintptr_t)gptr;
  v4u g0;
  g0.x = (unsigned)__builtin_amdgcn_readfirstlane(1);               // count=1, user D#
  g0.y = (unsigned)__builtin_amdgcn_readfirstlane((int)lds_addr);   // lds_addr
  g0.z = (unsigned)__builtin_amdgcn_readfirstlane((int)(unsigned)ga);
  g0.w = (unsigned)__builtin_amdgcn_readfirstlane(
      (int)(((unsigned)((ga >> 32) & 0x1FFFFFFull)) | (2u << 30))); // addr[56:32] | type=2
  int e = __builtin_amdgcn_readfirstlane((int)elems_f32);
  v8i g1;
  g1[0] = 0x00020000;                       // wg_mask=0, data_size=4B, no pad/iterate
  g1[1] = (e & 0xFFFF) << 16;               // tensor_dim0[15:0]
  g1[2] = ((e >> 16) & 0xFFFF) | (1 << 16); // tensor_dim0[31:16] | tensor_dim1=1
  g1[3] = (e & 0xFFFF) << 16;               // tile_dim0 = elems
  g1[4] = 1;                                // tile_dim1=1, tile_dim2=0
  g1[5] = e;                                // tensor_dim0_stride lo
  g1[6] = 0;                                // stride0 hi | stride1 lo
  g1[7] = 0;                                // stride1 hi
  v4i z4 = {0, 0, 0, 0};                    // groups 2/3 unused (<=2D tensor)
  v8i z8 = {0, 0, 0, 0, 0, 0, 0, 0};
  __builtin_amdgcn_tensor_load_to_lds(g0, g1, z4, z4, z8, 0);
}

__device__ __forceinline__ void load_row3(const float* __restrict__ tab, int idx,
                                          float& x, float& y, float& z) {
  const float* r = tab + (long long)idx * 3;
  x = r[0]; y = r[1]; z = r[2];   // contiguous -> global_load_b96 from L2
}

__global__ __launch_bounds__(TILE) void phong_kernel(
    const int*   __restrict__ p2f,      const float* __restrict__ bary,
    const float* __restrict__ verts,    const float* __restrict__ vnorm,
    const float* __restrict__ vcol,     const int*   __restrict__ faces,
    const float* __restrict__ light_loc,const float* __restrict__ light_amb,
    const float* __restrict__ light_dif,const float* __restrict__ light_spec,
    const float* __restrict__ mat_amb,  const float* __restrict__ mat_dif,
    const float* __restrict__ mat_spec, const float* __restrict__ shininess,
    const float* __restrict__ cam_pos,  const float* __restrict__ background,
    float* __restrict__ out, int hw, int total) {
  __shared__ int   sh_p2f[2][TILE];
  __shared__ float sh_bary[2][TILE * 3];

  const int tid   = threadIdx.x;
  const int wave  = tid >> 5;
  const int slice = wave * LANES;
  const int base  = blockIdx.x * (TILE * TILES_PER_BLOCK);

  // Kick off the DMA for tile 0 (per-wave private slice).
  if (base + slice < total) {
    tdm_load_1d(lds_byte_off(&sh_p2f[0][slice]),      p2f  + base + slice,            LANES);
    tdm_load_1d(lds_byte_off(&sh_bary[0][slice * 3]), bary + (long long)(base + slice) * 3,
                LANES * 3);
  }

  // Per-batch (uniform within a block: a block covers one n) + global constants.
  const int n = base / hw;
  const float lx = light_loc[n * 3 + 0], ly = light_loc[n * 3 + 1], lz = light_loc[n * 3 + 2];
  const float cx = cam_pos[n * 3 + 0],   cy = cam_pos[n * 3 + 1],   cz = cam_pos[n * 3 + 2];
  const float amb_r = mat_amb[0] * light_amb[0];
  const float amb_g = mat_amb[1] * light_amb[1];
  const float amb_b = mat_amb[2] * light_amb[2];
  const float dif_r = mat_dif[0] * light_dif[0];
  const float dif_g = mat_dif[1] * light_dif[1];
  const float dif_b = mat_dif[2] * light_dif[2];
  const float spc_r = mat_spec[0] * light_spec[0];
  const float spc_g = mat_spec[1] * light_spec[1];
  const float spc_b = mat_spec[2] * light_spec[2];
  const float shin  = shininess[0];
  const float bgr = background[0], bgg = background[1], bgb = background[2];

  for (int t = 0; t < TILES_PER_BLOCK; ++t) {
    const int buf = t & 1;
    const int tb  = base + t * TILE;

    if (t + 1 < TILES_PER_BLOCK) {
      const int nb  = buf ^ 1;
      const int ntb = tb + TILE;
      if (ntb + slice < total) {
        tdm_load_1d(lds_byte_off(&sh_p2f[nb][slice]),      p2f  + ntb + slice,            LANES);
        tdm_load_1d(lds_byte_off(&sh_bary[nb][slice * 3]), bary + (long long)(ntb + slice) * 3,
                    LANES * 3);
      }
      __builtin_amdgcn_s_wait_tensorcnt(2);   // current tile's 2 DMAs done (in-order)
    } else {
      __builtin_amdgcn_s_wait_tensorcnt(0);
    }

    const int   f  = sh_p2f[buf][tid];
    const float b0 = sh_bary[buf][tid * 3 + 0];
    const float b1 = sh_bary[buf][tid * 3 + 1];
    const float b2 = sh_bary[buf][tid * 3 + 2];

    const int fc = f < 0 ? 0 : f;            // reference: idx = max(p2f, 0)
    const long long fb = (long long)fc * 3;
    const int i0 = faces[fb + 0], i1 = faces[fb + 1], i2 = faces[fb + 2];

    float v0x, v0y, v0z, v1x, v1y, v1z, v2x, v2y, v2z;
    load_row3(verts, i0, v0x, v0y, v0z);
    load_row3(verts, i1, v1x, v1y, v1z);
    load_row3(verts, i2, v2x, v2y, v2z);
    float n0x, n0y, n0z, n1x, n1y, n1z, n2x, n2y, n2z;
    load_row3(vnorm, i0, n0x, n0y, n0z);
    load_row3(vnorm, i1, n1x, n1y, n1z);
    load_row3(vnorm, i2, n2x, n2y, n2z);
    float c0x, c0y, c0z, c1x, c1y, c1z, c2x, c2y, c2z;
    load_row3(vcol, i0, c0x, c0y, c0z);
    load_row3(vcol, i1, c1x, c1y, c1z);
    load_row3(vcol, i2, c2x, c2y, c2z);

    // Barycentric interpolation
    const float px = b0 * v0x + b1 * v1x + b2 * v2x;
    const float py = b0 * v0y + b1 * v1y + b2 * v2y;
    const float pz = b0 * v0z + b1 * v1z + b2 * v2z;
    float nx = b0 * n0x + b1 * n1x + b2 * n2x;
    float ny = b0 * n0y + b1 * n1y + b2 * n2y;
    float nz = b0 * n0z + b1 * n1z + b2 * n2z;
    const float tr = b0 * c0x + b1 * c1x + b2 * c2x;
    const float tg = b0 * c0y + b1 * c1y + b2 * c2y;
    const float tbc = b0 * c0z + b1 * c1z + b2 * c2z;

    // normalize(n)
    {
      const float nn  = sqrtf(nx * nx + ny * ny + nz * nz);
      const float inv = 1.0f / fmaxf(nn, 1e-6f);
      nx *= inv; ny *= inv; nz *= inv;
    }
    // to_light = normalize(light - p)
    float tlx = lx - px, tly = ly - py, tlz = lz - pz;
    {
      const float nn  = sqrtf(tlx * tlx + tly * tly + tlz * tlz);
      const float inv = 1.0f / fmaxf(nn, 1e-6f);
      tlx *= inv; tly *= inv; tlz *= inv;
    }
    const float cosang = nx * tlx + ny * tly + nz * tlz;
    const float cmax   = fmaxf(cosang, 0.0f);
    const float smask  = cosang > 0.0f ? 1.0f : 0.0f;
    const float cosm   = cosang * smask;
    // view = normalize(cam - p)
    float vx = cx - px, vy = cy - py, vz = cz - pz;
    {
      const float nn  = sqrtf(vx * vx + vy * vy + vz * vz);
      const float inv = 1.0f / fmaxf(nn, 1e-6f);
      vx *= inv; vy *= inv; vz *= inv;
    }
    // reflect = -to_light + 2*cosm*n
    const float rx = 2.0f * cosm * nx - tlx;
    const float ry = 2.0f * cosm * ny - tly;
    const float rz = 2.0f * cosm * nz - tlz;
    const float alpha = fmaxf(vx * rx + vy * ry + vz * rz, 0.0f) * smask;
    const float spec  = alpha > 0.0f ? __powf(alpha, shin) : 0.0f;

    const float cr = (amb_r + dif_r * cmax) * tr  + spc_r * spec;
    const float cg = (amb_g + dif_g * cmax) * tg  + spc_g * spec;
    const float cb = (amb_b + dif_b * cmax) * tbc + spc_b * spec;

    const bool is_bg = f < 0;
    v4f o;
    o.x = is_bg ? bgr : cr;
    o.y = is_bg ? bgg : cg;
    o.z = is_bg ? bgb : cb;
    o.w = is_bg ? 0.0f : 1.0f;

    const long long pix = (long long)tb + tid;
    if (pix < total) {
      __builtin_nontemporal_store(o, (v4f*)(out + pix * 4));  // b128 NT store
    }
  }
}

extern "C" void kernel_launch(void* const* d_in, const int* in_sizes, int n_in,
                              void* d_out, int out_size, void* d_ws, size_t ws_size,
                              hipStream_t stream) {
  (void)n_in; (void)d_ws; (void)ws_size; (void)out_size;
  const int*   p2f   = (const int*)  d_in[0];
  const float* bary  = (const float*)d_in[1];
  const float* verts = (const float*)d_in[2];
  const float* vnorm = (const float*)d_in[3];
  const float* vcol  = (const float*)d_in[4];
  const int*   faces = (const int*)  d_in[5];
  const float* lloc  = (const float*)d_in[6];
  const float* lamb  = (const float*)d_in[7];
  const float* ldif  = (const float*)d_in[8];
  const float* lspec = (const float*)d_in[9];
  const float* mamb  = (const float*)d_in[10];
  const float* mdif  = (const float*)d_in[11];
  const float* mspec = (const float*)d_in[12];
  const float* shin  = (const float*)d_in[13];
  const float* cam   = (const float*)d_in[14];
  const float* bg    = (const float*)d_in[15];
  float* out = (float*)d_out;

  const int total = in_sizes[0];          // N*H*W*K, K==1
  const int N     = in_sizes[6] / 3;      // light_location is (N,3)
  const int hw    = total / N;            // H*W (block never straddles an n)

  const int pixels_per_block = TILE * TILES_PER_BLOCK;
  const int blocks = (total + pixels_per_block - 1) / pixels_per_block;
  phong_kernel<<<blocks, TILE, 0, stream>>>(
      p2f, bary, verts, vnorm, vcol, faces, lloc, lamb, ldif, lspec,
      mamb, mdif, mspec, shin, cam, bg, out, hw, total);
}